// Tree_SSM_45990509806149
// MI455X (gfx1250) — compile-verified
//
#include <hip/hip_runtime.h>
#include <math.h>

typedef __attribute__((ext_vector_type(2))) float v2f;
typedef __attribute__((ext_vector_type(4))) float v4f;
typedef __attribute__((ext_vector_type(8))) float v8f;

constexpr int BB = 8, DD = 384, LL = 1024, RR = 24, CC = 26; // CC = R+2
constexpr int NE  = 992 + 992;   // horizontal + vertical grid edges = 1984
constexpr int NEP = 2048;        // padded (power of two) for bitonic sort

// ---------------------------------------------------------------------------
// Per-node feature norms:  norms[b,l] = ||x[b,:,l]|| + 1e-8
// ---------------------------------------------------------------------------
__global__ void node_norm_kernel(const float* __restrict__ x, float* __restrict__ norms) {
  int idx = blockIdx.x * blockDim.x + threadIdx.x;
  if (idx >= BB * LL) return;
  int b = idx / LL, l = idx % LL;
  float s = 0.f;
  for (int d = 0; d < DD; ++d) {
    float v = x[((size_t)b * DD + d) * LL + l];
    s += v * v;
  }
  norms[idx] = sqrtf(s) + 1e-8f;
}

// ---------------------------------------------------------------------------
// Edge cosine distances packed into stable-sortable 64-bit keys.
// One wave32 per edge; key = (sortable(dist) << 32) | edge_index.
// ---------------------------------------------------------------------------
__global__ __launch_bounds__(32) void edge_key_kernel(const float* __restrict__ x,
                                                      const float* __restrict__ norms,
                                                      unsigned long long* __restrict__ keys) {
  const int e = blockIdx.x % NEP;
  const int b = blockIdx.x / NEP;
  const int lane = threadIdx.x;
  if (e >= NE) { if (lane == 0) keys[(size_t)b * NEP + e] = ~0ull; return; }
  int u, v;
  if (e < 992) { int r = e / 31, c = e % 31; u = r * 32 + c; v = u + 1; } // ar[:, :-1] edges
  else         { u = e - 992; v = u + 32; }                              // ar[:-1, :] edges
  float s = 0.f;
  for (int d = lane; d < DD; d += 32)
    s += x[((size_t)b * DD + d) * LL + u] * x[((size_t)b * DD + d) * LL + v];
  for (int off = 16; off; off >>= 1) s += __shfl_xor(s, off, 32);
  if (lane == 0) {
    float dist = 1.f - s / (norms[b * LL + u] * norms[b * LL + v]);
    unsigned int bits = __float_as_uint(dist);
    bits = (bits & 0x80000000u) ? ~bits : (bits | 0x80000000u); // total-order float key
    keys[(size_t)b * NEP + e] = ((unsigned long long)bits << 32) | (unsigned)e;
  }
}

// ---------------------------------------------------------------------------
// One workgroup per batch: async-DMA keys into LDS, bitonic sort, serial
// Kruskal + BFS levels. LDS ~= 48 KB (WGP has 320 KB).
// ---------------------------------------------------------------------------
__global__ __launch_bounds__(1024) void tree_build_kernel(
    const unsigned long long* __restrict__ keysG,
    int* __restrict__ parentG, int* __restrict__ orderG,
    int* __restrict__ lstartG, int* __restrict__ numLevG) {
  __shared__ unsigned long long keys[NEP];
  __shared__ int uf[LL];
  __shared__ int par_s[LL];
  __shared__ int lev_s[LL];
  __shared__ int order_s[LL];
  __shared__ short adj[LL * 4];
  __shared__ unsigned char adjCnt[LL];
  __shared__ int lstart_s[LL + 2];

  const int b = blockIdx.x, t = threadIdx.x;

  // Async global->LDS copy of the 2048 keys (8 B per lane + mirrored copy at
  // +8192 B: INST_OFFSET is added to both the LDS and memory addresses).
  {
    const unsigned long long* gsrc = keysG + (size_t)b * NEP + t;
    unsigned lds0 = (unsigned)(size_t)&keys[t];
    asm volatile("global_load_async_to_lds_b64 %0, %1, off"
                 :: "v"(lds0), "v"(gsrc) : "memory");
    asm volatile("global_load_async_to_lds_b64 %0, %1, off offset:8192"
                 :: "v"(lds0), "v"(gsrc) : "memory");
  }
  uf[t] = t; lev_s[t] = -1; par_s[t] = 0; adjCnt[t] = 0; lstart_s[t] = 0;
  if (t == 0) { lstart_s[LL] = 0; lstart_s[LL + 1] = 0; }
  asm volatile("s_wait_asynccnt 0x0" ::: "memory");
  __syncthreads();

  // bitonic sort, ascending (stable tie-break via index in low bits)
  for (int k = 2; k <= NEP; k <<= 1) {
    for (int j = k >> 1; j > 0; j >>= 1) {
#pragma unroll
      for (int w = 0; w < 2; ++w) {
        int i = t + w * 1024;
        int p = i ^ j;
        if (p > i) {
          bool up = ((i & k) == 0);
          unsigned long long a = keys[i], c = keys[p];
          if ((a > c) == up) { keys[i] = c; keys[p] = a; }
        }
      }
      __syncthreads();
    }
  }

  if (t == 0) {
    // Kruskal (serial; ~2k union-find ops)
    int cnt = 0;
    for (int ei = 0; ei < NEP && cnt < LL - 1; ++ei) {
      int e = (int)(keys[ei] & 0xFFFFFFFFu);
      if (e >= NE) continue;
      int u, v;
      if (e < 992) { int r = e / 31, c = e % 31; u = r * 32 + c; v = u + 1; }
      else         { u = e - 992; v = u + 32; }
      int ru = u; while (uf[ru] != ru) { uf[ru] = uf[uf[ru]]; ru = uf[ru]; }
      int rv = v; while (uf[rv] != rv) { uf[rv] = uf[uf[rv]]; rv = uf[rv]; }
      if (ru != rv) {
        uf[ru] = rv;
        adj[u * 4 + adjCnt[u]++] = (short)v;
        adj[v * 4 + adjCnt[v]++] = (short)u;
        ++cnt;
      }
    }
    // BFS from node 0, recording level boundaries
    order_s[0] = 0; lev_s[0] = 0;
    lstart_s[0] = 0; lstart_s[1] = 1;
    int s = 0, eL = 1, nl = 1;
    while (eL > s) {
      int ne = eL;
      for (int i = s; i < eL; ++i) {
        int u = order_s[i];
        int c = adjCnt[u];
        for (int a = 0; a < c; ++a) {
          int v = adj[u * 4 + a];
          if (lev_s[v] < 0) { lev_s[v] = 1; par_s[v] = u; order_s[ne++] = v; }
        }
      }
      if (ne > eL) { ++nl; lstart_s[nl] = ne; }
      s = eL; eL = ne;
    }
    numLevG[b] = nl;
    for (int i = 0; i <= nl; ++i) lstartG[b * (LL + 2) + i] = lstart_s[i];
  }
  __syncthreads();
  parentG[b * LL + t] = par_s[t];
  orderG[b * LL + t]  = order_s[t];
}

// ---------------------------------------------------------------------------
// f32 GEMM on the matrix engine: V_WMMA_F32_16X16X4_F32.
// One wave computes a 16x64 C strip = 4 tiles sharing one A fragment.
// Column-interleaved tile mapping: tile t owns columns nt + 4*mi + t, so per
// K-step the loads are 1 x b64 (A pair) + 2 x b128 (both B rows, all 4 tiles)
// feeding 4 WMMAs; C is written with 8 x b128. GEMM columns are permutation-
// invariant so the interleaving is transparent.
// A fragment (16x4): lanes 0-15 hold K={0,1}, lanes 16-31 hold K={2,3}.
// C VGPR i: lanes 0-15 -> row i, lanes 16-31 -> row i+8.  K % 4 == 0, N % 64 == 0.
// ---------------------------------------------------------------------------
__global__ __launch_bounds__(32) void wmma_gemm_f32_16x16x4(
    const float* __restrict__ A, const float* __restrict__ B, float* __restrict__ C,
    int M, int N, int K, long sA, long sB, long sC) {
  const int b = blockIdx.z;
  A += (size_t)b * sA; B += (size_t)b * sB; C += (size_t)b * sC;
  const int lane = threadIdx.x & 31;
  const int mt = blockIdx.y << 4;
  const int nt = blockIdx.x << 6;          // 64 columns per wave
  const int half = lane >> 4;              // K-pair selector
  const int mi   = lane & 15;              // A row / B-C lane column group
  const int row  = mt + mi;
  const float as = (row < M) ? 1.f : 0.f;  // branch-free M tail (loop-invariant)
  const int rowc = (row < M) ? row : (M - 1);
  const float* Ap = A + (size_t)rowc * K + (half << 1);
  const float* Bp = B + (size_t)(half << 1) * N + nt + (mi << 2);
  v8f acc0 = {}, acc1 = {}, acc2 = {}, acc3 = {};
  for (int k = 0; k < K; k += 4) {
    v2f a = *(const v2f*)Ap;
    a.x *= as; a.y *= as;
    v4f b0 = *(const v4f*)Bp;        // row ka,   4 tile columns
    v4f b1 = *(const v4f*)(Bp + N);  // row ka+1, 4 tile columns
    if (k + 4 < K) __builtin_prefetch(Bp + 4 * N, 0, 1);
    v2f f0 = {b0.x, b1.x};
    v2f f1 = {b0.y, b1.y};
    v2f f2 = {b0.z, b1.z};
    v2f f3 = {b0.w, b1.w};
    acc0 = __builtin_amdgcn_wmma_f32_16x16x4_f32(false, a, false, f0, (short)0, acc0, false, false);
    acc1 = __builtin_amdgcn_wmma_f32_16x16x4_f32(false, a, false, f1, (short)0, acc1, false, false);
    acc2 = __builtin_amdgcn_wmma_f32_16x16x4_f32(false, a, false, f2, (short)0, acc2, false, false);
    acc3 = __builtin_amdgcn_wmma_f32_16x16x4_f32(false, a, false, f3, (short)0, acc3, false, false);
    Ap += 4;
    Bp += (size_t)4 * N;
  }
#pragma unroll
  for (int i = 0; i < 8; ++i) {
    const int r = mt + i + (half << 3);
    if (r < M) {
      v4f o = {acc0[i], acc1[i], acc2[i], acc3[i]};
      *(v4f*)(C + (size_t)r * N + nt + (mi << 2)) = o;
    }
  }
}

// ---------------------------------------------------------------------------
// deltaA = exp(softplus(dts + bias) * (-exp(A_log)))   (in place over dts)
// BX     = softplus(...) * Bsv * x                     (-> S buffer)
// ---------------------------------------------------------------------------
__global__ void deltaA_BX_kernel(float* __restrict__ dts, float* __restrict__ Sbuf,
                                 const float* __restrict__ x, const float* __restrict__ xdbl,
                                 const float* __restrict__ dtb, const float* __restrict__ Al) {
  size_t idx = (size_t)blockIdx.x * blockDim.x + threadIdx.x;
  if (idx >= (size_t)BB * DD * LL) return;
  int l = (int)(idx % LL);
  int d = (int)((idx / LL) % DD);
  int b = (int)(idx / ((size_t)DD * LL));
  float raw = dts[idx] + dtb[d];
  float sp  = raw > 20.f ? raw : log1pf(expf(raw));
  dts[idx]  = expf(sp * (-expf(Al[d])));                    // deltaA (edge weight w)
  float Bsv = xdbl[((size_t)b * CC + RR) * LL + l];         // row R of x_dbl
  Sbuf[idx] = sp * Bsv * x[idx];                            // BX (scan init)
}

// ---------------------------------------------------------------------------
// Tree scans: one workgroup per (batch, 64-channel chunk); levels are
// sequential with in-block barriers (device-side dynamic level count — no
// data-dependent host loop, graph-capture safe). Up: scatter-add to parents.
// Down: h[nd] = S[nd] + w[nd]*(h[pa] - w[nd]*S[nd]), in place over S.
// ---------------------------------------------------------------------------
__global__ __launch_bounds__(256) void tree_scan_kernel(
    float* __restrict__ S, const float* __restrict__ w,
    const int* __restrict__ order, const int* __restrict__ parent,
    const int* __restrict__ lstart, const int* __restrict__ numLev) {
  const int b = blockIdx.y;
  const int c0 = blockIdx.x * 64;
  const int nl = numLev[b];
  S += (size_t)b * DD * LL; w += (size_t)b * DD * LL;
  order += b * LL; parent += b * LL; lstart += b * (LL + 2);
  for (int lev = nl - 1; lev >= 1; --lev) {
    const int s = lstart[lev], n = lstart[lev + 1] - s;
    for (int p = threadIdx.x; p < n * 64; p += 256) {
      const int i = p >> 6, c = c0 + (p & 63);
      const int nd = order[s + i], pa = parent[nd];
      atomicAdd(&S[(size_t)c * LL + pa], w[(size_t)c * LL + nd] * S[(size_t)c * LL + nd]);
    }
    __threadfence(); __syncthreads();
  }
  for (int lev = 1; lev < nl; ++lev) {
    const int s = lstart[lev], n = lstart[lev + 1] - s;
    for (int p = threadIdx.x; p < n * 64; p += 256) {
      const int i = p >> 6, c = c0 + (p & 63);
      const int nd = order[s + i], pa = parent[nd];
      float wn = w[(size_t)c * LL + nd];
      float sn = S[(size_t)c * LL + nd];
      float hp = S[(size_t)c * LL + pa];
      S[(size_t)c * LL + nd] = sn + wn * (hp - wn * sn);
    }
    __threadfence(); __syncthreads();
  }
}

// ---------------------------------------------------------------------------
// Per-(b,l): layer_norm(h)*Csv + Ds*x, then layer_norm -> out (B,H,W,D).
// ---------------------------------------------------------------------------
__device__ __forceinline__ float blockSum(float v, float* red) {
  int t = threadIdx.x;
  red[t] = v; __syncthreads();
  for (int s = 64; s > 0; s >>= 1) { if (t < s) red[t] += red[t + s]; __syncthreads(); }
  float r = red[0]; __syncthreads();
  return r;
}

__global__ __launch_bounds__(128) void output_kernel(
    const float* __restrict__ hbuf, const float* __restrict__ x,
    const float* __restrict__ xdbl, const float* __restrict__ Ds,
    const float* __restrict__ hw, const float* __restrict__ hb,
    const float* __restrict__ ow, const float* __restrict__ ob,
    float* __restrict__ out) {
  const int b = blockIdx.x / LL, l = blockIdx.x % LL;
  const int t = threadIdx.x;
  __shared__ float red[128];
  float hv[3], yv[3];
#pragma unroll
  for (int k = 0; k < 3; ++k)
    hv[k] = hbuf[((size_t)b * DD + (t + k * 128)) * LL + l];
  float mu = blockSum(hv[0] + hv[1] + hv[2], red) * (1.f / DD);
  float q = 0.f;
#pragma unroll
  for (int k = 0; k < 3; ++k) { float df = hv[k] - mu; q += df * df; }
  float inv = rsqrtf(blockSum(q, red) * (1.f / DD) + 1e-5f);
  float Csv = xdbl[((size_t)b * CC + RR + 1) * LL + l];
#pragma unroll
  for (int k = 0; k < 3; ++k) {
    int d = t + k * 128;
    float o = (hv[k] - mu) * inv * hw[d] + hb[d];
    yv[k] = o * Csv + Ds[d] * x[((size_t)b * DD + d) * LL + l];
  }
  float mu2 = blockSum(yv[0] + yv[1] + yv[2], red) * (1.f / DD);
  float q2 = 0.f;
#pragma unroll
  for (int k = 0; k < 3; ++k) { float df = yv[k] - mu2; q2 += df * df; }
  float inv2 = rsqrtf(blockSum(q2, red) * (1.f / DD) + 1e-5f);
#pragma unroll
  for (int k = 0; k < 3; ++k) {
    int d = t + k * 128;
    out[((size_t)b * LL + l) * DD + d] = (yv[k] - mu2) * inv2 * ow[d] + ob[d];
  }
}

// ---------------------------------------------------------------------------
extern "C" void kernel_launch(void* const* d_in, const int* in_sizes, int n_in,
                              void* d_out, int out_size, void* d_ws, size_t ws_size,
                              hipStream_t stream) {
  const float* x   = (const float*)d_in[0];
  const float* xpw = (const float*)d_in[1];
  const float* dtw = (const float*)d_in[2];
  const float* dtb = (const float*)d_in[3];
  const float* Al  = (const float*)d_in[4];
  const float* Dsp = (const float*)d_in[5];
  const float* hw  = (const float*)d_in[6];
  const float* hb  = (const float*)d_in[7];
  const float* ow  = (const float*)d_in[8];
  const float* ob  = (const float*)d_in[9];
  float* out = (float*)d_out;

  char* ws = (char*)d_ws;
  size_t off = 0;
  auto alloc = [&](size_t bytes) -> void* {
    void* p = ws + off;
    off += (bytes + 255) & ~(size_t)255;
    return p;
  };
  float* dtsBuf = (float*)alloc((size_t)BB * DD * LL * 4);   // dts -> deltaA (w)
  float* Sbuf   = (float*)alloc((size_t)BB * DD * LL * 4);   // BX -> S -> h
  float* xdbl   = (float*)alloc((size_t)BB * CC * LL * 4);
  float* norms  = (float*)alloc((size_t)BB * LL * 4);
  unsigned long long* keys = (unsigned long long*)alloc((size_t)BB * NEP * 8);
  int* parent = (int*)alloc((size_t)BB * LL * 4);
  int* order  = (int*)alloc((size_t)BB * LL * 4);
  int* lstart = (int*)alloc((size_t)BB * (LL + 2) * 4);
  int* numLev = (int*)alloc((size_t)BB * 4);

  // tree construction path
  node_norm_kernel<<<(BB * LL + 255) / 256, 256, 0, stream>>>(x, norms);
  edge_key_kernel<<<BB * NEP, 32, 0, stream>>>(x, norms, keys);
  tree_build_kernel<<<BB, 1024, 0, stream>>>(keys, parent, order, lstart, numLev);

  // projection path (WMMA f32): x_dbl = Wp @ X ; dts = Wdt @ dts_r
  wmma_gemm_f32_16x16x4<<<dim3(LL / 64, 2, BB), 32, 0, stream>>>(
      xpw, x, xdbl, CC, LL, DD, 0L, (long)DD * LL, (long)CC * LL);
  wmma_gemm_f32_16x16x4<<<dim3(LL / 64, DD / 16, BB), 32, 0, stream>>>(
      dtw, xdbl, dtsBuf, DD, LL, RR, 0L, (long)CC * LL, (long)DD * LL);

  deltaA_BX_kernel<<<(int)(((size_t)BB * DD * LL + 255) / 256), 256, 0, stream>>>(
      dtsBuf, Sbuf, x, xdbl, dtb, Al);

  tree_scan_kernel<<<dim3(DD / 64, BB), 256, 0, stream>>>(
      Sbuf, dtsBuf, order, parent, lstart, numLev);

  output_kernel<<<BB * LL, 128, 0, stream>>>(
      Sbuf, x, xdbl, Dsp, hw, hb, ow, ob, out);
}